// Model_54451595378718
// MI455X (gfx1250) — compile-verified
//
#include <hip/hip_runtime.h>

typedef unsigned int v4u __attribute__((ext_vector_type(4)));
typedef int          v4i __attribute__((ext_vector_type(4)));
typedef int          v8i __attribute__((ext_vector_type(8)));
typedef float        v4f __attribute__((ext_vector_type(4)));   // clang vector: OK for nontemporal builtins

// Problem constants (from reference): x is (8192, 8192) f32, split at H=4096.
constexpr int MROWS = 8192;
constexpr int NCOLS = 8192;
constexpr int HALF  = NCOLS / 2;          // 4096
constexpr int BLOCK = 256;                // 8 wave32s per row
constexpr int VEC   = 4;                  // float4 granularity
constexpr int CHUNKS = HALF / (BLOCK * VEC); // 4 float4 chunks per half per thread
constexpr int NWAVES = BLOCK / 32;

#if defined(__HIP_DEVICE_COMPILE__) && __has_builtin(__builtin_amdgcn_tensor_load_to_lds)
#define USE_TDM 1
#endif

__global__ __launch_bounds__(BLOCK)
void swiglu_rowquant_kernel(const float* __restrict__ x,
                            const float* __restrict__ weight_scale,
                            const float* __restrict__ activation_scale,
                            const float* __restrict__ quant_scale,
                            float* __restrict__ out)
{
    // Full row (front half then back half) staged in LDS: 32 KB of 320 KB/WGP.
    __shared__ float smem[2 * HALF];
    __shared__ float wave_max[NWAVES];

    const int row = blockIdx.x;
    const int tid = threadIdx.x;

    // Workgroup-relative LDS byte address of the staging buffer.
    const unsigned lds_base = (unsigned)(unsigned long long)(&smem[0]);

#ifdef USE_TDM
    // ---- Tensor Data Mover: one descriptor moves the whole 32 KB row ----
    if (tid == 0) {
        const unsigned long long gaddr =
            (unsigned long long)x + (unsigned long long)row * (unsigned long long)(NCOLS * 4);

        // D# group 0: count=1 | lds_addr | global_addr[56:0] | type=2 (ISA 8.3)
        v4u g0 = (v4u){ 1u,
                        lds_base,
                        (unsigned)gaddr,
                        ((unsigned)(gaddr >> 32) & 0x01FFFFFFu) | 0x80000000u };

        // D# group 1 (ISA 8.4): data_size=4B; tensor_dim0=8192; tensor_dim1=1;
        // tile_dim0=8192; tile_dim1=1; tensor_dim0_stride=8192; tensor_dim1_stride=8192.
        v8i g1 = (v8i){ (int)(2u << 16),       // data_size=2 (4 bytes), mask=0
                        (int)0x20000000u,      // tensor_dim0[15:0]=8192 -> bits 63:48
                        (int)0x00010000u,      // tensor_dim1[15:0]=1    -> bits 95:80
                        (int)0x20000000u,      // tile_dim0=8192         -> bits 127:112
                        (int)0x00000001u,      // tile_dim1=1, tile_dim2=0
                        (int)8192,             // tensor_dim0_stride[31:0]
                        (int)0x20000000u,      // tensor_dim1_stride[15:0]=8192 -> bits 223:208
                        0 };

        v4i gz4 = (v4i){0, 0, 0, 0};           // groups 2/3 unused (2-D tensor)

#if __clang_major__ >= 23
        v8i gz8 = (v8i){0, 0, 0, 0, 0, 0, 0, 0};
        __builtin_amdgcn_tensor_load_to_lds(g0, g1, gz4, gz4, gz8, 0);
#else
        __builtin_amdgcn_tensor_load_to_lds(g0, g1, gz4, gz4, 0);
#endif
        __builtin_amdgcn_s_wait_tensorcnt(0);
    }
#else
    // ---- Fallback: per-lane async global->LDS b128 loads (ASYNCcnt path) ----
    const unsigned row_byte = (unsigned)row * (unsigned)NCOLS * 4u;
    const unsigned long long xbase = (unsigned long long)x;
#pragma unroll
    for (int v = 0; v < CHUNKS; ++v) {
        const unsigned e      = (unsigned)(v * BLOCK + tid) * (unsigned)VEC;
        const unsigned lds_f  = lds_base + e * 4u;
        const unsigned lds_g  = lds_base + (unsigned)HALF * 4u + e * 4u;
        const unsigned gofs_f = row_byte + e * 4u;
        const unsigned gofs_g = row_byte + (unsigned)HALF * 4u + e * 4u;
        asm volatile("global_load_async_to_lds_b128 %0, %1, %2"
                     :: "v"(lds_f), "v"(gofs_f), "s"(xbase) : "memory");
        asm volatile("global_load_async_to_lds_b128 %0, %1, %2"
                     :: "v"(lds_g), "v"(gofs_g), "s"(xbase) : "memory");
    }
    asm volatile("s_wait_asynccnt 0" ::: "memory");
#endif

    // Uniform scalars (latency overlapped with the DMA).
    const float k  = weight_scale[0] * activation_scale[row];
    const float qs = quant_scale[0];

    __syncthreads();   // LDS now fully populated for all waves

    // Single pass: SwiGLU into registers + local absmax.
    float sval[CHUNKS * VEC];
    float amax = 0.0f;
#pragma unroll
    for (int v = 0; v < CHUNKS; ++v) {
        const int e = (v * BLOCK + tid) * VEC;
        const v4f f = *reinterpret_cast<const v4f*>(&smem[e]);          // lin (front)
        const v4f g = *reinterpret_cast<const v4f*>(&smem[HALF + e]);   // glu (back)
#pragma unroll
        for (int j = 0; j < 4; ++j) {
            const float lin = f[j] * k;
            const float glu = g[j] * k;
            const float sig = 1.0f / (1.0f + __expf(-glu));
            const float s   = lin * glu * sig * qs;
            sval[v * 4 + j] = s;
            amax = fmaxf(amax, fabsf(s));
        }
    }

    // wave32 absmax reduction (warpSize == 32 on gfx1250).
#pragma unroll
    for (int off = 16; off > 0; off >>= 1)
        amax = fmaxf(amax, __shfl_xor(amax, off, 32));

    const int lane = tid & 31;
    const int wid  = tid >> 5;
    if (lane == 0) wave_max[wid] = amax;
    __syncthreads();

    float rmax = wave_max[0];
#pragma unroll
    for (int i = 1; i < NWAVES; ++i) rmax = fmaxf(rmax, wave_max[i]);

    const float scale = rmax * (1.0f / 127.0f);
    // scale==0 => all s are 0 => emit zeros; inv=0 reproduces reference's where().
    const float inv = (scale > 0.0f) ? (1.0f / scale) : 0.0f;

    // Quantize; stream out with non-temporal b128 stores (no reuse, keep L2 clean).
    float* __restrict__ orow = out + (size_t)row * (size_t)HALF;
#pragma unroll
    for (int v = 0; v < CHUNKS; ++v) {
        const int e = (v * BLOCK + tid) * VEC;
        v4f q;
#pragma unroll
        for (int j = 0; j < 4; ++j) {
            float t = rintf(sval[v * 4 + j] * inv);   // round-half-to-even, like jnp.round
            t = fminf(fmaxf(t, -128.0f), 127.0f);
            q[j] = t;
        }
        __builtin_nontemporal_store(q, reinterpret_cast<v4f*>(&orow[e]));
    }

    if (tid == 0)
        __builtin_nontemporal_store(scale, &out[(size_t)MROWS * (size_t)HALF + (size_t)row]);
}

extern "C" void kernel_launch(void* const* d_in, const int* in_sizes, int n_in,
                              void* d_out, int out_size, void* d_ws, size_t ws_size,
                              hipStream_t stream) {
    (void)in_sizes; (void)n_in; (void)out_size; (void)d_ws; (void)ws_size;
    const float* x  = (const float*)d_in[0];
    const float* ws = (const float*)d_in[1];
    const float* as = (const float*)d_in[2];
    const float* qs = (const float*)d_in[3];
    float* out = (float*)d_out;

    swiglu_rowquant_kernel<<<dim3(MROWS), dim3(BLOCK), 0, stream>>>(x, ws, as, qs, out);
}